// CTCLoss_13804024889448
// MI455X (gfx1250) — compile-verified
//
#include <hip/hip_runtime.h>
#include <stdint.h>

#define NT      2000
#define NB      256
#define NF      5
#define NS      500
#define TILE    128
#define NTILES  ((NT + TILE - 1) / TILE)   // 16 (last tile = 80 steps)
#define NEGF    (-1e30f)

// ---------------- CDNA5 async global->LDS helpers ----------------
#if defined(__HIP_DEVICE_COMPILE__) && __has_builtin(__builtin_amdgcn_global_load_async_to_lds_b32)
#define HAVE_ASYNC_LDS 1
typedef __attribute__((address_space(1))) int as1_int;   // global pointee (builtin takes int*)
typedef __attribute__((address_space(3))) int as3_int;   // LDS pointee
__device__ __forceinline__ void async_copy_b32(const float* g, float* l) {
    // integer round-trip to materialize AS1 / AS3 pointers (low 32 bits of a
    // generic shared pointer are the LDS byte address on gfx1250)
    __builtin_amdgcn_global_load_async_to_lds_b32(
        (as1_int*)(unsigned long long)(uintptr_t)g,
        (as3_int*)(unsigned int)(uintptr_t)l,
        0, 0);
}
__device__ __forceinline__ void wait_async() {
#if __has_builtin(__builtin_amdgcn_s_wait_asynccnt)
    __builtin_amdgcn_s_wait_asynccnt(0);
#else
    asm volatile("s_wait_asynccnt 0x0" ::: "memory");
#endif
}
#else
#define HAVE_ASYNC_LDS 0
__device__ __forceinline__ void async_copy_b32(const float* g, float* l) { *l = *g; }
__device__ __forceinline__ void wait_async() {}
#endif

// logaddexp with native transcendentals: abs error of log(1+e) for tiny e is
// negligible relative to the accumulated forward scores.
__device__ __forceinline__ float lae(float a, float b) {
    float m = fmaxf(a, b);
    float d = fminf(a, b) - m;        // <= 0, inputs always finite (-1e30 floor)
    float e = __expf(d);              // v_exp_f32
    return m + __logf(1.0f + e);      // v_log_f32
}

__global__ __launch_bounds__(256) void ctc_scan_kernel(const float* __restrict__ x,
                                                       const int*   __restrict__ seqs,
                                                       const int*   __restrict__ seqlens,
                                                       float*       __restrict__ out) {
    __shared__ float s_x[2][TILE * NF];   // double-buffered x tiles (2 * 2560 B)
    __shared__ float s_fwd[2][257];       // double-buffered neighbor values, [0] = -inf sentinel
    __shared__ float s_fin[2 * 256];      // final fwd vector for the output gather

    const int tid = threadIdx.x;
    const int b   = blockIdx.x;

    // Positions owned by this thread: p0 = 2*tid, p1 = 2*tid+1.
    // Move into position p uses seqs[b][p-1].
    const int pm0  = 2 * tid - 1;
    const int pm1  = 2 * tid;
    const int idx0 = (pm0 >= 0 && pm0 < NS) ? seqs[b * NS + pm0] : 0;
    const int idx1 = (pm1 < NS)             ? seqs[b * NS + pm1] : 0;

    float f0 = (tid == 0) ? 0.0f : NEGF;  // fwd[p0]
    float f1 = NEGF;                      // fwd[p1]

    s_fwd[0][tid + 1] = NEGF;
    if (tid == 0) { s_fwd[0][0] = NEGF; s_fwd[1][0] = NEGF; }

    // Prefetch tile 0: element e -> (t_local = e/5, feature = e%5)
    for (int e2 = tid; e2 < TILE * NF; e2 += 256) {
        const int tl = e2 / NF, f = e2 - tl * NF;
        const float* g = x + ((size_t)tl * NB + b) * NF + f;
        async_copy_b32(g, &s_x[0][e2]);
    }
    wait_async();
    __syncthreads();

    int rb = 0;                            // read buffer for s_fwd
    for (int jt = 0; jt < NTILES; ++jt) {
        const int cb = jt & 1;
        // Kick off next tile's async loads; they complete while we run the
        // 128 serial DP steps of the current tile (tracked by ASYNCcnt).
        if (jt + 1 < NTILES) {
            for (int e2 = tid; e2 < TILE * NF; e2 += 256) {
                const int tl = e2 / NF, f = e2 - tl * NF;
                int tg = (jt + 1) * TILE + tl;
                if (tg >= NT) tg = NT - 1;                 // clamp (values unused)
                const float* g = x + ((size_t)tg * NB + b) * NF + f;
                async_copy_b32(g, &s_x[cb ^ 1][e2]);
            }
        }

        const float* xt = s_x[cb];
        const int kmax = (NT - jt * TILE < TILE) ? (NT - jt * TILE) : TILE;
        for (int k = 0; k < kmax; ++k) {
            const float x4   = xt[k * NF + 4];      // stay score (broadcast)
            const float xm0  = xt[k * NF + idx0];   // 4 banks, conflict-free
            const float xm1  = xt[k * NF + idx1];
            const float left = s_fwd[rb][tid];      // old fwd[p0-1] (sentinel -inf at tid==0)

            const float n0 = lae(xm0 + left, x4 + f0);   // new fwd[p0]
            const float n1 = lae(xm1 + f0,   x4 + f1);   // new fwd[p1] (old f0 in-register)

            s_fwd[rb ^ 1][tid + 1] = n1;            // publish for right neighbor
            f0 = n0; f1 = n1;
            rb ^= 1;
            __syncthreads();                        // single barrier per DP step
        }

        wait_async();                               // next tile landed in LDS
        __syncthreads();
    }

    // Gather output: out[b] = -fwd[seqlens[b]] / (nt * SHARP)
    s_fin[2 * tid]     = f0;
    s_fin[2 * tid + 1] = f1;
    __syncthreads();
    if (tid == 0) {
        const int sl = seqlens[b];                  // 0..500, all stored positions valid
        out[b] = -s_fin[sl] * (1.0f / (float)NT);
    }
}

extern "C" void kernel_launch(void* const* d_in, const int* in_sizes, int n_in,
                              void* d_out, int out_size, void* d_ws, size_t ws_size,
                              hipStream_t stream) {
    (void)in_sizes; (void)n_in; (void)out_size; (void)d_ws; (void)ws_size;
    const float* x        = (const float*)d_in[0];
    const int*   seqs     = (const int*)d_in[1];
    const int*   seqlens  = (const int*)d_in[2];
    float*       out      = (float*)d_out;

    dim3 grid(NB);      // one workgroup per batch row
    dim3 block(256);    // 8 wave32 waves; each thread owns 2 DP positions
    hipLaunchKernelGGL(ctc_scan_kernel, grid, block, 0, stream, x, seqs, seqlens, out);
}